// NetVLAD_12498354831502
// MI455X (gfx1250) — compile-verified
//
#include <hip/hip_runtime.h>

typedef float v2f __attribute__((ext_vector_type(2)));
typedef float v8f __attribute__((ext_vector_type(8)));

#define NB   32
#define DD   512
#define KK   64
#define PP   1024
#define EPSF 1e-12f

// -------------------------------------------------------------------------
// Kernel 1: logits = conv_w @ x + conv_b (per 16-pixel tile), softmax over K.
// Output layout: a_pair[n][p/2][k] = {a[p][k], a[p+1][k]}  (v2f elements),
// so the aggregation GEMM can fetch its whole B operand with one b64 load.
// Block = 128 threads (4 waves). Wave w owns k-rows [16w, 16w+16).
// -------------------------------------------------------------------------
__global__ __launch_bounds__(128) void k_assign(const float* __restrict__ x,
                                                const float* __restrict__ conv_w,
                                                const float* __restrict__ conv_b,
                                                v2f* __restrict__ a_pair) {
    __shared__ float sm[16 * KK];    // [p_local][k]
    __shared__ float pmax[16 * 8];   // partial max, 8 groups per pixel
    __shared__ float psum[16 * 8];   // partial exp-sum
    const int n    = blockIdx.x >> 6;             // P/16 = 64 tiles per image
    const int p0   = (blockIdx.x & 63) << 4;
    const int tid  = threadIdx.x;
    const int wave = tid >> 5;
    const int lane = tid & 31;
    const int k0   = wave << 4;
    const int hi   = (lane >= 16) ? 1 : 0;
    const int l15  = lane & 15;
    const float* xb = x + (size_t)n * DD * PP;

    v8f c = {0.f, 0.f, 0.f, 0.f, 0.f, 0.f, 0.f, 0.f};

    // A = conv_w tile [16 k x 4 d]: lanes 0-15 -> K=d+0,d+1 ; lanes 16-31 -> d+2,d+3
    // B = x tile      [4 d x 16 p]: vgpr0 rows d+0/d+2, vgpr1 rows d+1/d+3
    for (int d = 0; d < DD; d += 4) {
        v2f a = *(const v2f*)(conv_w + (size_t)(k0 + l15) * DD + d + (hi ? 2 : 0));
        const int dr = d + (hi ? 2 : 0);
        v2f b;
        b.x = xb[(size_t)dr * PP + p0 + l15];
        b.y = xb[(size_t)(dr + 1) * PP + p0 + l15];
        c = __builtin_amdgcn_wmma_f32_16x16x4_f32(false, a, false, b,
                                                  (short)0, c, false, false);
    }

    // C layout: c[r] holds (M = r + 8*hi, N = l15). Add bias, park in LDS as [p][k].
#pragma unroll
    for (int r = 0; r < 8; ++r) {
        const int k = k0 + r + (hi ? 8 : 0);
        sm[l15 * KK + k] = c[r] + conv_b[k];
    }
    __syncthreads();

    // Parallel softmax over K=64: 8 threads per pixel, 8 clusters each.
    const int p = tid >> 3;
    const int g = tid & 7;
    float m = -3.0e38f;
#pragma unroll
    for (int j = 0; j < 8; ++j) m = fmaxf(m, sm[p * KK + g * 8 + j]);
    pmax[p * 8 + g] = m;
    __syncthreads();
    float M = pmax[p * 8];
#pragma unroll
    for (int j = 1; j < 8; ++j) M = fmaxf(M, pmax[p * 8 + j]);
    float s = 0.f;
#pragma unroll
    for (int j = 0; j < 8; ++j) {
        const float e = __expf(sm[p * KK + g * 8 + j] - M);
        sm[p * KK + g * 8 + j] = e;
        s += e;
    }
    psum[p * 8 + g] = s;
    __syncthreads();
    float S = psum[p * 8];
#pragma unroll
    for (int j = 1; j < 8; ++j) S += psum[p * 8 + j];
    const float inv = 1.0f / S;
#pragma unroll
    for (int j = 0; j < 8; ++j) sm[p * KK + g * 8 + j] *= inv;
    __syncthreads();

    // Coalesced pair-interleaved store: 8 pixel-pairs x 64 k = 512 v2f.
    v2f* ao = a_pair + ((size_t)n * (PP / 2) + (p0 >> 1)) * KK;
    for (int i = tid; i < 8 * KK; i += 128) {
        const int ppl = i >> 6;      // local pixel pair
        const int k   = i & 63;
        v2f v;
        v.x = sm[(2 * ppl) * KK + k];
        v.y = sm[(2 * ppl + 1) * KK + k];
        ao[(size_t)ppl * KK + k] = v;
    }
}

// -------------------------------------------------------------------------
// Kernel 2: asum[n][k] = sum_p a[n][k][p]   (k contiguous, b64 loads)
// -------------------------------------------------------------------------
__global__ __launch_bounds__(64) void k_asum(const v2f* __restrict__ a_pair,
                                             float* __restrict__ asum) {
    const int n = blockIdx.x;
    const int k = threadIdx.x;
    const v2f* ap = a_pair + (size_t)n * (PP / 2) * KK + k;
    float s = 0.f;
    for (int pp = 0; pp < PP / 2; ++pp) {
        const v2f v = ap[(size_t)pp * KK];
        s += v.x + v.y;
    }
    asum[n * KK + k] = s;
}

// -------------------------------------------------------------------------
// Kernel 3: vladT tile GEMM:  vlad[k][d] = sum_p a[k][p]*x[d][p] - asum[k]*c[k][d]
// Computed as M=d (rows), N=k (cols): A (x) is float2-contiguous per lane,
// B (a_pair) is one b64 per lane per step thanks to the pair-interleave.
// Grid (N, K/16, D/64), block = 128 threads (4 waves, one 16-wide d tile each).
// -------------------------------------------------------------------------
__global__ __launch_bounds__(128) void k_vlad(const float* __restrict__ x,
                                              const v2f* __restrict__ a_pair,
                                              const float* __restrict__ asum,
                                              const float* __restrict__ centroids,
                                              float* __restrict__ vraw) {
    const int n    = blockIdx.x;
    const int k0   = blockIdx.y << 4;
    const int tid  = threadIdx.x;
    const int wave = tid >> 5;
    const int lane = tid & 31;
    const int d0   = (blockIdx.z << 6) + (wave << 4);
    const int hi   = (lane >= 16) ? 1 : 0;
    const int l15  = lane & 15;
    const float* xb = x + (size_t)n * DD * PP;
    const v2f* ab = a_pair + (size_t)n * (PP / 2) * KK;

    v8f c = {0.f, 0.f, 0.f, 0.f, 0.f, 0.f, 0.f, 0.f};

    // A = x tile [16 d x 4 p]: lane row d0+l15, K cols {p,p+1} (lo) / {p+2,p+3} (hi)
    // B = a tile [4 p x 16 k]: b64 gives {a[pr][k], a[pr+1][k]} = vgpr0/vgpr1
    for (int p = 0; p < PP; p += 4) {
        const int pr  = p + (hi ? 2 : 0);
        v2f a = *(const v2f*)(xb + (size_t)(d0 + l15) * PP + pr);
        v2f b = ab[(size_t)((p >> 1) + hi) * KK + k0 + l15];
        c = __builtin_amdgcn_wmma_f32_16x16x4_f32(false, a, false, b,
                                                  (short)0, c, false, false);
    }

    const int k = k0 + l15;
    const float as = asum[n * KK + k];
    float* vo = vraw + (size_t)n * KK * DD;
#pragma unroll
    for (int r = 0; r < 8; ++r) {
        const int d = d0 + r + (hi ? 8 : 0);
        vo[(size_t)k * DD + d] = c[r] - as * centroids[(size_t)k * DD + d];
    }
}

// -------------------------------------------------------------------------
// Kernel 4: intra-normalize each (k) row over d, then global L2 normalize.
// One block per image n.
// -------------------------------------------------------------------------
__global__ __launch_bounds__(256) void k_norm(const float* __restrict__ vraw,
                                              float* __restrict__ out) {
    __shared__ float red[256];
    __shared__ float tsc[KK];
    const int n   = blockIdx.x;
    const int tid = threadIdx.x;
    const float* vp = vraw + (size_t)n * KK * DD;

    // Row sum-of-squares: 4 threads per k row.
    const int k   = tid >> 2;
    const int sub = tid & 3;
    float s = 0.f;
    for (int d = sub; d < DD; d += 4) {
        const float v = vp[(size_t)k * DD + d];
        s += v * v;
    }
    red[tid] = s;
    __syncthreads();

    float t = 0.f;
    if (tid < KK)
        t = red[tid * 4] + red[tid * 4 + 1] + red[tid * 4 + 2] + red[tid * 4 + 3];
    __syncthreads();
    if (tid < KK) {
        const float nrm = sqrtf(t);
        const float sc  = 1.0f / fmaxf(nrm, EPSF);
        const float nn  = nrm * sc;      // norm of the normalized row (1 or tiny)
        red[tid] = nn * nn;
        tsc[tid] = sc;
    }
    __syncthreads();
    if (tid == 0) {
        float g = 0.f;
        for (int i = 0; i < KK; ++i) g += red[i];
        red[0] = 1.0f / fmaxf(sqrtf(g), EPSF);
    }
    __syncthreads();

    const float gs = red[0];
    float* op = out + (size_t)n * KK * DD;
    for (int i = tid; i < KK * DD; i += 256) {
        op[i] = vp[i] * tsc[i >> 9] * gs;    // i>>9 == i / DD
    }
}

// -------------------------------------------------------------------------
extern "C" void kernel_launch(void* const* d_in, const int* in_sizes, int n_in,
                              void* d_out, int out_size, void* d_ws, size_t ws_size,
                              hipStream_t stream) {
    const float* x         = (const float*)d_in[0];   // [32,512,32,32]
    const float* centroids = (const float*)d_in[1];   // [64,512]
    const float* conv_w    = (const float*)d_in[2];   // [64,512]
    const float* conv_b    = (const float*)d_in[3];   // [64]
    float* out = (float*)d_out;

    // Workspace layout (floats): a_pair [N*P*K] | asum [N*K] | vraw [N*K*D]
    v2f*   a_pair = (v2f*)d_ws;                                  // N*(P/2)*K v2f
    float* asum   = (float*)d_ws + (size_t)NB * PP * KK;
    float* vraw   = asum + (size_t)NB * KK;
    (void)in_sizes; (void)n_in; (void)out_size; (void)ws_size;

    k_assign<<<dim3(NB * (PP / 16)), 128, 0, stream>>>(x, conv_w, conv_b, a_pair);
    k_asum  <<<dim3(NB), 64, 0, stream>>>(a_pair, asum);
    k_vlad  <<<dim3(NB, KK / 16, DD / 64), 128, 0, stream>>>(x, a_pair, asum, centroids, vraw);
    k_norm  <<<dim3(NB), 256, 0, stream>>>(vraw, out);
}